// FNN2d_FewShot_Spatial_3753801417196
// MI455X (gfx1250) — compile-verified
//
#include <hip/hip_runtime.h>

typedef float v2f __attribute__((ext_vector_type(2)));
typedef float v8f __attribute__((ext_vector_type(8)));
typedef float f4  __attribute__((ext_vector_type(4)));
typedef int   v4i __attribute__((ext_vector_type(4)));

// x layout: (B=8, 519, 128, 128) fp32. Channels: [0,64)=query, [64,512)=demo_X (J=7,C=64), [512,519)=demo_Y
#define CH_PLANE 16384            // 128*128
#define B_STRIDE (519 * 16384)
#define LSTR 72                   // LDS token-stride per channel row:
                                  //   half-wave reads ch+2 -> +144 words == +16 banks (mod 64): conflict-free

// ---- CDNA5 async global->LDS path (ASYNCcnt), with safe fallback -----------
#if __has_builtin(__builtin_amdgcn_global_load_async_to_lds_b128) && \
    __has_builtin(__builtin_amdgcn_s_wait_asynccnt)
#define USE_ASYNC 1
#else
#define USE_ASYNC 0
#endif

typedef __attribute__((address_space(1))) v4i as1_v4i;   // global int4
typedef __attribute__((address_space(3))) v4i as3_v4i;   // LDS int4

__device__ __forceinline__ void copy16_g2l(const float* gp, float* lp) {
#if USE_ASYNC
  __builtin_amdgcn_global_load_async_to_lds_b128((as1_v4i*)(void*)gp, (as3_v4i*)lp, 0, 0);
#else
  *(f4*)lp = *(const f4*)gp;
#endif
}

#if USE_ASYNC
#define WAIT_ASYNC(n) __builtin_amdgcn_s_wait_asynccnt(n)
#else
#define WAIT_ASYNC(n) do { } while (0)
#endif

// ---------------------------------------------------------------------------
// Kernel 1: windowed attention, one workgroup per (b, wh, ww) window.
// 4 waves; wave rt owns query rows [rt*16, rt*16+16). Online softmax over the
// 7 demo chunks (one demo image = one 64-token chunk), double-buffered through
// LDS with async global->LDS copies. Writes out[p] = 0.5 * yw[p].
// ---------------------------------------------------------------------------
__global__ __launch_bounds__(128) void attn_win_kernel(const float* __restrict__ x,
                                                       float* __restrict__ out) {
  __shared__ float Qs[64 * LSTR];          // Qs[ch][token]
  __shared__ float Ds[2][64 * LSTR];       // double-buffered demo chunk

  const int win  = blockIdx.x;             // 0..2047
  const int b    = win >> 8;
  const int w2   = win & 255;
  const int wh   = w2 >> 4;
  const int ww   = w2 & 15;
  const int tid  = threadIdx.x;
  const int wave = tid >> 5;
  const int lane = tid & 31;
  const int l16  = lane & 15;
  const int half = lane >> 4;

  const size_t base = (size_t)b * B_STRIDE + (size_t)(wh * 8) * 128 + (size_t)(ww * 8);

  // Per-thread staging slice: 8 x 16-byte transfers per 16KB chunk.
  // idx -> (ch = idx>>4, q = idx&15); tokens q*4..q*4+3 are 4 consecutive pixels.
  int st_ch[8], st_q[8];
#pragma unroll
  for (int it = 0; it < 8; ++it) {
    const int idx = it * 128 + tid;
    st_ch[it] = idx >> 4;
    st_q[it]  = idx & 15;
  }

  // ---- prologue: async-stage Q window and demo chunk 0 ----------------------
#pragma unroll
  for (int it = 0; it < 8; ++it) {
    const int ch = st_ch[it], q = st_q[it];
    copy16_g2l(x + base + (size_t)ch * CH_PLANE + (q >> 1) * 128 + (q & 1) * 4,
               &Qs[ch * LSTR + q * 4]);
  }
#pragma unroll
  for (int it = 0; it < 8; ++it) {
    const int ch = st_ch[it], q = st_q[it];
    copy16_g2l(x + base + (size_t)(64 + ch) * CH_PLANE + (q >> 1) * 128 + (q & 1) * 4,
               &Ds[0][ch * LSTR + q * 4]);
  }

  // online-softmax state: this lane (redundantly, per 16-lane column group)
  // tracks the 8 rows l = rt*16 + half*8 + v.
  float mrow[8], Zrow[8], Yrow[8];
#pragma unroll
  for (int v = 0; v < 8; ++v) { mrow[v] = -3.0e38f; Zrow[v] = 0.0f; Yrow[v] = 0.0f; }

  const int rt   = wave;                   // row tile 0..3
  const int chA  = half * 2;               // per-lane K offset inside a 4-wide K step
  const int tokA = rt * 16 + l16;          // A-fragment token this lane reads

  for (int j = 0; j < 7; ++j) {
    // ---- issue next chunk while this one is consumed ------------------------
    if (j < 6) {
      const size_t dbase = base + (size_t)(64 + (j + 1) * 64) * CH_PLANE;
      float* dbuf = &Ds[(j + 1) & 1][0];
#pragma unroll
      for (int it = 0; it < 8; ++it) {
        const int ch = st_ch[it], q = st_q[it];
        copy16_g2l(x + dbase + (size_t)ch * CH_PLANE + (q >> 1) * 128 + (q & 1) * 4,
                   &dbuf[ch * LSTR + q * 4]);
      }
      WAIT_ASYNC(8);    // chunk j (and Q on j==0) complete; chunk j+1 in flight
    } else {
      WAIT_ASYNC(0);
    }
    __syncthreads();

    const float* __restrict__ D = &Ds[j & 1][0];

    // ---- S tile block: 16 rows x 64 cols via f32 WMMA -----------------------
    v8f acc0 = {}, acc1 = {}, acc2 = {}, acc3 = {};
#pragma unroll
    for (int k0 = 0; k0 < 64; k0 += 4) {
      const int r0 = (k0 + chA) * LSTR;
      const int r1 = r0 + LSTR;
      v2f a;
      a.x = Qs[r0 + tokA];
      a.y = Qs[r1 + tokA];
      v2f b0, b1, b2, b3;
      b0.x = D[r0 + 0  + l16];  b0.y = D[r1 + 0  + l16];
      b1.x = D[r0 + 16 + l16];  b1.y = D[r1 + 16 + l16];
      b2.x = D[r0 + 32 + l16];  b2.y = D[r1 + 32 + l16];
      b3.x = D[r0 + 48 + l16];  b3.y = D[r1 + 48 + l16];
      acc0 = __builtin_amdgcn_wmma_f32_16x16x4_f32(false, a, false, b0, (short)0, acc0, false, false);
      acc1 = __builtin_amdgcn_wmma_f32_16x16x4_f32(false, a, false, b1, (short)0, acc1, false, false);
      acc2 = __builtin_amdgcn_wmma_f32_16x16x4_f32(false, a, false, b2, (short)0, acc2, false, false);
      acc3 = __builtin_amdgcn_wmma_f32_16x16x4_f32(false, a, false, b3, (short)0, acc3, false, false);
    }

    // ---- wy values for this chunk: one per column this lane owns ------------
    const size_t ybase = base + (size_t)(512 + j) * CH_PLANE;
    float wy0, wy1, wy2, wy3;
    {
      const int s0 = l16, s1 = 16 + l16, s2 = 32 + l16, s3 = 48 + l16;
      wy0 = x[ybase + (s0 >> 3) * 128 + (s0 & 7)];
      wy1 = x[ybase + (s1 >> 3) * 128 + (s1 & 7)];
      wy2 = x[ybase + (s2 >> 3) * 128 + (s2 & 7)];
      wy3 = x[ybase + (s3 >> 3) * 128 + (s3 & 7)];
    }

    // ---- online softmax update per owned row --------------------------------
#pragma unroll
    for (int v = 0; v < 8; ++v) {
      float cmax = fmaxf(fmaxf(acc0[v], acc1[v]), fmaxf(acc2[v], acc3[v]));
#pragma unroll
      for (int off = 8; off >= 1; off >>= 1)
        cmax = fmaxf(cmax, __shfl_xor(cmax, off, 32));
      const float mnew  = fmaxf(mrow[v], cmax);
      const float scale = __expf(mrow[v] - mnew);
      const float e0 = __expf(acc0[v] - mnew);
      const float e1 = __expf(acc1[v] - mnew);
      const float e2 = __expf(acc2[v] - mnew);
      const float e3 = __expf(acc3[v] - mnew);
      float pz = e0 + e1 + e2 + e3;
      float py = e0 * wy0 + e1 * wy1 + e2 * wy2 + e3 * wy3;
#pragma unroll
      for (int off = 8; off >= 1; off >>= 1) {
        pz += __shfl_xor(pz, off, 32);
        py += __shfl_xor(py, off, 32);
      }
      Zrow[v] = Zrow[v] * scale + pz;
      Yrow[v] = Yrow[v] * scale + py;
      mrow[v] = mnew;
    }
    __syncthreads();   // buffer j&1 may be overwritten next iteration
  }

  // ---- write 0.5 * yw: lanes 0 and 16 of each wave own 8 rows each ----------
  if (l16 == 0) {
#pragma unroll
    for (int v = 0; v < 8; ++v) {
      const int l  = rt * 16 + half * 8 + v;
      const int hh = wh * 8 + (l >> 3);
      const int wp = ww * 8 + (l & 7);
      out[(size_t)b * CH_PLANE + hh * 128 + wp] = 0.5f * (Yrow[v] / Zrow[v]);
    }
  }
}

// ---------------------------------------------------------------------------
// Kernel 2: per-pixel MLP  y = tanh(q @ W1 + b1) @ W2 + b2.
// One wave per 16-pixel tile; 8 feature tiles x 16 K-steps of f32 WMMA.
// Accumulates out[p] += 0.5 * y[p].
// ---------------------------------------------------------------------------
__global__ __launch_bounds__(128) void fc_kernel(const float* __restrict__ x,
                                                 const float* __restrict__ w1,
                                                 const float* __restrict__ b1,
                                                 const float* __restrict__ w2,
                                                 const float* __restrict__ b2,
                                                 float* __restrict__ out) {
  const int tid  = threadIdx.x;
  const int wave = tid >> 5;
  const int lane = tid & 31;
  const int l16  = lane & 15;
  const int half = lane >> 4;

  const int tileM = blockIdx.x * 4 + wave;   // 8192 tiles of 16 pixels
  const int P0    = tileM * 16;
  const int bb    = P0 >> 14;                // / 16384 pixels per batch image
  const int off0  = P0 & 16383;

  const size_t qbase = (size_t)bb * B_STRIDE + (size_t)off0 + (size_t)l16;
  const int chA = half * 2;

  v8f acc[8] = {{}, {}, {}, {}, {}, {}, {}, {}};
#pragma unroll
  for (int k0 = 0; k0 < 64; k0 += 4) {
    v2f a;
    const int ch = k0 + chA;
    a.x = x[qbase + (size_t)ch * CH_PLANE];
    a.y = x[qbase + (size_t)(ch + 1) * CH_PLANE];
#pragma unroll
    for (int t = 0; t < 8; ++t) {
      v2f bf;
      const int f = t * 16 + l16;
      bf.x = w1[ch * 128 + f];
      bf.y = w1[(ch + 1) * 128 + f];
      acc[t] = __builtin_amdgcn_wmma_f32_16x16x4_f32(false, a, false, bf, (short)0, acc[t], false, false);
    }
  }

  // bias + tanh + 128->1 projection (per-lane partial, then 16-lane butterfly)
  float b1v[8], w2v[8];
#pragma unroll
  for (int t = 0; t < 8; ++t) {
    const int f = t * 16 + l16;
    b1v[t] = b1[f];
    w2v[t] = w2[f];
  }
  const float bias2 = b2[0];

  float part[8];
#pragma unroll
  for (int v = 0; v < 8; ++v) {
    float p = 0.0f;
#pragma unroll
    for (int t = 0; t < 8; ++t)
      p += tanhf(acc[t][v] + b1v[t]) * w2v[t];
    part[v] = p;
  }
#pragma unroll
  for (int v = 0; v < 8; ++v) {
#pragma unroll
    for (int off = 8; off >= 1; off >>= 1)
      part[v] += __shfl_xor(part[v], off, 32);
  }

  if (l16 == 0) {
#pragma unroll
    for (int v = 0; v < 8; ++v) {
      const size_t p = (size_t)P0 + (size_t)(half * 8 + v);
      out[p] = out[p] + 0.5f * (part[v] + bias2);
    }
  }
}

// ---------------------------------------------------------------------------
extern "C" void kernel_launch(void* const* d_in, const int* in_sizes, int n_in,
                              void* d_out, int out_size, void* d_ws, size_t ws_size,
                              hipStream_t stream) {
  (void)in_sizes; (void)n_in; (void)out_size; (void)d_ws; (void)ws_size;
  const float* x  = (const float*)d_in[0];
  const float* w1 = (const float*)d_in[1];
  const float* b1 = (const float*)d_in[2];
  const float* w2 = (const float*)d_in[3];
  const float* b2 = (const float*)d_in[4];
  float* out = (float*)d_out;

  // kernel 1 writes 0.5*yw; kernel 2 (stream-ordered after) adds 0.5*fc(x).
  attn_win_kernel<<<dim3(2048), dim3(128), 0, stream>>>(x, out);
  fc_kernel<<<dim3(2048), dim3(128), 0, stream>>>(x, w1, b1, w2, b2, out);
}